// LocalTrafficModel_20693152432563
// MI455X (gfx1250) — compile-verified
//
#include <hip/hip_runtime.h>
#include <math.h>

// Problem constants
static constexpr int NB   = 8;       // batches
static constexpr int NN   = 2048;    // nodes per batch
static constexpr int DIN  = 64;
static constexpr int DH   = 128;
static constexpr int DG   = 32;
static constexpr int DOUT = 12;
static constexpr int NENV = 4;
static constexpr int BN   = NB * NN; // 16384 rows

#define ACT_NONE 0
#define ACT_RELU 1
#define ACT_TANH 2

typedef __attribute__((ext_vector_type(2))) float v2f;
typedef __attribute__((ext_vector_type(8))) float v8f;

// CDNA5 fp32 WMMA: D(16x16,f32) = A(16x4,f32) x B(4x16,f32) + C
__device__ __forceinline__ v8f wmma4(v2f a, v2f b, v8f c) {
  return __builtin_amdgcn_wmma_f32_16x16x4_f32(false, a, false, b, (short)0, c,
                                               false, false);
}

// CDNA5 hardware tanh (TRANS32 unit, co-executes with matrix pipe)
__device__ __forceinline__ float fast_tanh(float x) {
#if __has_builtin(__builtin_amdgcn_tanhf)
  return __builtin_amdgcn_tanhf(x);
#else
  float r;
  asm("v_tanh_f32 %0, %1" : "=v"(r) : "v"(x));
  return r;
#endif
}

// Generic flat pointer to LDS (low 32 bits of generic address = LDS offset)
__device__ __forceinline__ unsigned ldsoff(const void* p) {
  return (unsigned)(size_t)p;
}

// CDNA5 async copy global -> LDS, 16 bytes per lane, tracked by ASYNCcnt
__device__ __forceinline__ void async_ld16(unsigned lds_addr, const float* g) {
  asm volatile("global_load_async_to_lds_b128 %0, %1, off"
               :: "v"(lds_addr), "v"(g)
               : "memory");
}
#define WAIT_ASYNC(n) asm volatile("s_wait_asynccnt " n ::: "memory")

// ---------------------------------------------------------------------------
// Generic row-linear: Y[r,0:OUT] = act(X[r,:K] @ W[OUT,K]^T + bias)
// One wave per 16-row tile; loops over 16-col tiles; K multiple of 4.
// ---------------------------------------------------------------------------
__global__ __launch_bounds__(32)
void lin_wmma_kernel(const float* __restrict__ X, const float* __restrict__ W,
                     const float* __restrict__ bias, float* __restrict__ Y,
                     int K, int OUT, int act) {
  const int lane = threadIdx.x;
  const int j = lane & 15;
  const int g = lane >> 4;
  const int r0 = blockIdx.x * 16;
  const int ctiles = (OUT + 15) >> 4;

  for (int ct = 0; ct < ctiles; ++ct) {
    const int c0 = ct << 4;
    const int jc = (c0 + j < OUT) ? (c0 + j) : (OUT - 1);  // clamp partial tiles
    v8f acc = {};
    for (int k0 = 0; k0 < K; k0 += 4) {
      const int ka = k0 + 2 * g;
      v2f a, b;
      a.x = X[(size_t)(r0 + j) * K + ka];
      a.y = X[(size_t)(r0 + j) * K + ka + 1];
      b.x = W[(size_t)jc * K + ka];
      b.y = W[(size_t)jc * K + ka + 1];
      acc = wmma4(a, b, acc);
    }
    const int col = c0 + j;
    if (col < OUT) {
      const float bv = bias[col];
#pragma unroll
      for (int v = 0; v < 8; ++v) {
        float val = acc[v] + bv;
        if (act == ACT_RELU)      val = fmaxf(val, 0.0f);
        else if (act == ACT_TANH) val = fast_tanh(val);
        Y[(size_t)(r0 + v + 8 * g) * OUT + col] = val;
      }
    }
  }
}

// ---------------------------------------------------------------------------
// Fused GRU pass 1: block = 4 waves = 64 n-rows of one batch. Per m-step the
// block stages M1/M2/val_u/val_r m-tiles in LDS with double-buffered async
// copies; each wave builds its adj tile = relu(tanh(t1[n,m]-t1[m,n]^T)) via
// WMMA (+LDS transpose) and accumulates adj @ val_u, adj @ val_r.
// ---------------------------------------------------------------------------
__global__ __launch_bounds__(128)
void gru_ur_kernel(const float* __restrict__ M1, const float* __restrict__ M2,
                   const float* __restrict__ VU, const float* __restrict__ VR,
                   float* __restrict__ UG, float* __restrict__ RG) {
  __shared__ __align__(16) float sVU[2][16 * DH];
  __shared__ __align__(16) float sVR[2][16 * DH];
  __shared__ __align__(16) float sM1[2][16 * DG];
  __shared__ __align__(16) float sM2[2][16 * DG];
  __shared__ float sT[4][16 * 16];
  __shared__ float sAdj[4][16 * 16];

  const int tid  = threadIdx.x;
  const int wave = tid >> 5;
  const int lane = tid & 31;
  const int j = lane & 15;
  const int g = lane >> 4;
  const int b  = blockIdx.y;
  const int n0 = (blockIdx.x << 6) + (wave << 4);

  const float* M1b = M1 + (size_t)b * NN * DG;
  const float* M2b = M2 + (size_t)b * NN * DG;
  const float* Ub  = VU + (size_t)b * NN * DH;
  const float* Rb  = VR + (size_t)b * NN * DH;

  // 10 async ops per buffer: 4 (val_u) + 4 (val_r) + 1 (M1) + 1 (M2)
  auto issue = [&](int bb, int m0) {
    const unsigned bvu = ldsoff(&sVU[bb][0]);
    const unsigned bvr = ldsoff(&sVR[bb][0]);
    const unsigned bm1 = ldsoff(&sM1[bb][0]);
    const unsigned bm2 = ldsoff(&sM2[bb][0]);
#pragma unroll
    for (int q = 0; q < 4; ++q) {
      const int idx = q * 128 + tid;                 // 512 float4 = 16x128
      async_ld16(bvu + idx * 16, Ub + (size_t)(m0 + (idx >> 5)) * DH + (idx & 31) * 4);
    }
#pragma unroll
    for (int q = 0; q < 4; ++q) {
      const int idx = q * 128 + tid;
      async_ld16(bvr + idx * 16, Rb + (size_t)(m0 + (idx >> 5)) * DH + (idx & 31) * 4);
    }
    async_ld16(bm1 + tid * 16, M1b + (size_t)(m0 + (tid >> 3)) * DG + (tid & 7) * 4);
    async_ld16(bm2 + tid * 16, M2b + (size_t)(m0 + (tid >> 3)) * DG + (tid & 7) * 4);
  };

  issue(0, 0);

  // Register-resident n-tile fragments of M1 (A) and M2 (B)
  v2f a_n[8], b_n[8];
#pragma unroll
  for (int kk = 0; kk < 8; ++kk) {
    const int ka = kk * 4 + 2 * g;
    a_n[kk].x = M1b[(size_t)(n0 + j) * DG + ka];
    a_n[kk].y = M1b[(size_t)(n0 + j) * DG + ka + 1];
    b_n[kk].x = M2b[(size_t)(n0 + j) * DG + ka];
    b_n[kk].y = M2b[(size_t)(n0 + j) * DG + ka + 1];
  }

  v8f accU[8], accR[8];
#pragma unroll
  for (int t = 0; t < 8; ++t) { accU[t] = (v8f){}; accR[t] = (v8f){}; }

  for (int mi = 0; mi < NN / 16; ++mi) {
    const int cur = mi & 1;
    __syncthreads();                 // next buffer free to overwrite
    if (mi + 1 < NN / 16) {
      issue(cur ^ 1, (mi + 1) * 16);
      WAIT_ASYNC("0xa");             // in-order: current buffer complete
    } else {
      WAIT_ASYNC("0x0");
    }
    __syncthreads();                 // current buffer visible to all waves

    v8f tnm = {}, tmn = {};
#pragma unroll
    for (int kk = 0; kk < 8; ++kk) {
      const int ka = kk * 4 + 2 * g;
      v2f a_m, b_m;
      a_m.x = sM1[cur][j * DG + ka];
      a_m.y = sM1[cur][j * DG + ka + 1];
      b_m.x = sM2[cur][j * DG + ka];
      b_m.y = sM2[cur][j * DG + ka + 1];
      tnm = wmma4(a_n[kk], b_m, tnm);  // t1[n-tile, m-tile]
      tmn = wmma4(a_m, b_n[kk], tmn);  // t1[m-tile, n-tile]
    }
    // Transpose t1[m,n] tile through wave-private LDS (DS in-order per wave)
#pragma unroll
    for (int v = 0; v < 8; ++v) sT[wave][(v + 8 * g) * 16 + j] = tmn[v];
#pragma unroll
    for (int v = 0; v < 8; ++v) {
      const int i = v + 8 * g;
      float d = fast_tanh(tnm[v] - sT[wave][j * 16 + i]);
      sAdj[wave][i * 16 + j] = d > 0.0f ? d : 0.0f;
    }
    // adj tile as A-operand, shared across 8 col tiles per k-step
#pragma unroll
    for (int kk = 0; kk < 16; kk += 4) {
      const int kb = kk + 2 * g;
      v2f aa;
      aa.x = sAdj[wave][j * 16 + kb];
      aa.y = sAdj[wave][j * 16 + kb + 1];
#pragma unroll
      for (int ct = 0; ct < 8; ++ct) {
        const int col = ct * 16 + j;
        v2f bu, br;
        bu.x = sVU[cur][kb * DH + col];
        bu.y = sVU[cur][(kb + 1) * DH + col];
        br.x = sVR[cur][kb * DH + col];
        br.y = sVR[cur][(kb + 1) * DH + col];
        accU[ct] = wmma4(aa, bu, accU[ct]);
        accR[ct] = wmma4(aa, br, accR[ct]);
      }
    }
  }

#pragma unroll
  for (int ct = 0; ct < 8; ++ct) {
    const int col = ct * 16 + j;
#pragma unroll
    for (int v = 0; v < 8; ++v) {
      const size_t row = (size_t)b * NN + n0 + v + 8 * g;
      UG[row * DH + col] = 1.0f / (1.0f + __expf(-accU[ct][v]));
      RG[row * DH + col] = 1.0f / (1.0f + __expf(-accR[ct][v]));
    }
  }
}

// ---------------------------------------------------------------------------
// Fused GRU pass 2: same adj recomputation; accumulate adj @ val_c; epilogue
// c = tanh(.), h_new = u*h_prev + (1-u)*c.
// ---------------------------------------------------------------------------
__global__ __launch_bounds__(128)
void gru_c_kernel(const float* __restrict__ M1, const float* __restrict__ M2,
                  const float* __restrict__ VC, const float* __restrict__ UG,
                  const float* __restrict__ hprev, float* __restrict__ hnew) {
  __shared__ __align__(16) float sVC[2][16 * DH];
  __shared__ __align__(16) float sM1[2][16 * DG];
  __shared__ __align__(16) float sM2[2][16 * DG];
  __shared__ float sT[4][16 * 16];
  __shared__ float sAdj[4][16 * 16];

  const int tid  = threadIdx.x;
  const int wave = tid >> 5;
  const int lane = tid & 31;
  const int j = lane & 15;
  const int g = lane >> 4;
  const int b  = blockIdx.y;
  const int n0 = (blockIdx.x << 6) + (wave << 4);

  const float* M1b = M1 + (size_t)b * NN * DG;
  const float* M2b = M2 + (size_t)b * NN * DG;
  const float* Cb  = VC + (size_t)b * NN * DH;

  // 6 async ops per buffer: 4 (val_c) + 1 (M1) + 1 (M2)
  auto issue = [&](int bb, int m0) {
    const unsigned bvc = ldsoff(&sVC[bb][0]);
    const unsigned bm1 = ldsoff(&sM1[bb][0]);
    const unsigned bm2 = ldsoff(&sM2[bb][0]);
#pragma unroll
    for (int q = 0; q < 4; ++q) {
      const int idx = q * 128 + tid;
      async_ld16(bvc + idx * 16, Cb + (size_t)(m0 + (idx >> 5)) * DH + (idx & 31) * 4);
    }
    async_ld16(bm1 + tid * 16, M1b + (size_t)(m0 + (tid >> 3)) * DG + (tid & 7) * 4);
    async_ld16(bm2 + tid * 16, M2b + (size_t)(m0 + (tid >> 3)) * DG + (tid & 7) * 4);
  };

  issue(0, 0);

  v2f a_n[8], b_n[8];
#pragma unroll
  for (int kk = 0; kk < 8; ++kk) {
    const int ka = kk * 4 + 2 * g;
    a_n[kk].x = M1b[(size_t)(n0 + j) * DG + ka];
    a_n[kk].y = M1b[(size_t)(n0 + j) * DG + ka + 1];
    b_n[kk].x = M2b[(size_t)(n0 + j) * DG + ka];
    b_n[kk].y = M2b[(size_t)(n0 + j) * DG + ka + 1];
  }

  v8f accC[8];
#pragma unroll
  for (int t = 0; t < 8; ++t) accC[t] = (v8f){};

  for (int mi = 0; mi < NN / 16; ++mi) {
    const int cur = mi & 1;
    __syncthreads();
    if (mi + 1 < NN / 16) {
      issue(cur ^ 1, (mi + 1) * 16);
      WAIT_ASYNC("0x6");
    } else {
      WAIT_ASYNC("0x0");
    }
    __syncthreads();

    v8f tnm = {}, tmn = {};
#pragma unroll
    for (int kk = 0; kk < 8; ++kk) {
      const int ka = kk * 4 + 2 * g;
      v2f a_m, b_m;
      a_m.x = sM1[cur][j * DG + ka];
      a_m.y = sM1[cur][j * DG + ka + 1];
      b_m.x = sM2[cur][j * DG + ka];
      b_m.y = sM2[cur][j * DG + ka + 1];
      tnm = wmma4(a_n[kk], b_m, tnm);
      tmn = wmma4(a_m, b_n[kk], tmn);
    }
#pragma unroll
    for (int v = 0; v < 8; ++v) sT[wave][(v + 8 * g) * 16 + j] = tmn[v];
#pragma unroll
    for (int v = 0; v < 8; ++v) {
      const int i = v + 8 * g;
      float d = fast_tanh(tnm[v] - sT[wave][j * 16 + i]);
      sAdj[wave][i * 16 + j] = d > 0.0f ? d : 0.0f;
    }
#pragma unroll
    for (int kk = 0; kk < 16; kk += 4) {
      const int kb = kk + 2 * g;
      v2f aa;
      aa.x = sAdj[wave][j * 16 + kb];
      aa.y = sAdj[wave][j * 16 + kb + 1];
#pragma unroll
      for (int ct = 0; ct < 8; ++ct) {
        const int col = ct * 16 + j;
        v2f bc;
        bc.x = sVC[cur][kb * DH + col];
        bc.y = sVC[cur][(kb + 1) * DH + col];
        accC[ct] = wmma4(aa, bc, accC[ct]);
      }
    }
  }

#pragma unroll
  for (int ct = 0; ct < 8; ++ct) {
    const int col = ct * 16 + j;
#pragma unroll
    for (int v = 0; v < 8; ++v) {
      const size_t row = (size_t)b * NN + n0 + v + 8 * g;
      const size_t idx = row * DH + col;
      const float c = fast_tanh(accC[ct][v]);
      const float u = UG[idx];
      hnew[idx] = u * hprev[idx] + (1.0f - u) * c;
    }
  }
}

// ---------------------------------------------------------------------------
// Elementwise helpers: build ci = [h_inv, h_prev]; later overwrite the second
// half with r*h_prev to form c_in (first half h_inv is unchanged).
// ---------------------------------------------------------------------------
__global__ __launch_bounds__(256)
void concat_ci_kernel(const float* __restrict__ hinv,
                      const float* __restrict__ hprev, float* __restrict__ ci) {
  const int idx = blockIdx.x * 256 + threadIdx.x;  // over BN*DH
  const int r = idx >> 7, c = idx & 127;
  ci[(size_t)r * 256 + c]       = hinv[idx];
  ci[(size_t)r * 256 + 128 + c] = hprev[idx];
}

__global__ __launch_bounds__(256)
void cin_update_kernel(const float* __restrict__ rg,
                       const float* __restrict__ hprev, float* __restrict__ ci) {
  const int idx = blockIdx.x * 256 + threadIdx.x;
  const int r = idx >> 7, c = idx & 127;
  ci[(size_t)r * 256 + 128 + c] = rg[idx] * hprev[idx];
}

// ---------------------------------------------------------------------------
extern "C" void kernel_launch(void* const* d_in, const int* in_sizes, int n_in,
                              void* d_out, int out_size, void* d_ws, size_t ws_size,
                              hipStream_t stream) {
  const float* x      = (const float*)d_in[0];
  const float* h_prev = (const float*)d_in[1];
  const float* ei_w1  = (const float*)d_in[2];
  const float* ei_b1  = (const float*)d_in[3];
  const float* ei_w2  = (const float*)d_in[4];
  const float* ei_b2  = (const float*)d_in[5];
  const float* es_w1  = (const float*)d_in[6];
  const float* es_b1  = (const float*)d_in[7];
  const float* es_w2  = (const float*)d_in[8];
  const float* es_b2  = (const float*)d_in[9];
  const float* env_w  = (const float*)d_in[10];
  const float* env_b  = (const float*)d_in[11];
  const float* g1_w   = (const float*)d_in[12];
  const float* g1_b   = (const float*)d_in[13];
  const float* g2_w   = (const float*)d_in[14];
  const float* g2_b   = (const float*)d_in[15];
  const float* wu_w   = (const float*)d_in[16];
  const float* wu_b   = (const float*)d_in[17];
  const float* wr_w   = (const float*)d_in[18];
  const float* wr_b   = (const float*)d_in[19];
  const float* wc_w   = (const float*)d_in[20];
  const float* wc_b   = (const float*)d_in[21];
  const float* pr_w   = (const float*)d_in[22];
  const float* pr_b   = (const float*)d_in[23];

  // Output layout: prediction | env_logits | h_inv | h_new
  float* out  = (float*)d_out;
  float* pred = out;
  float* envl = pred + (size_t)BN * DOUT;
  float* hinv = envl + (size_t)BN * NENV;
  float* hnew = hinv + (size_t)BN * DH;

  // Workspace carve-up (~64 MB, stays resident in 192 MB L2)
  float* ws  = (float*)d_ws;
  float* T1  = ws;  ws += (size_t)BN * DH;      // enc hidden (reused), later r-gate
  float* HS  = ws;  ws += (size_t)BN * DH;      // h_spu
  float* Mm1 = ws;  ws += (size_t)BN * DG;
  float* Mm2 = ws;  ws += (size_t)BN * DG;
  float* CI  = ws;  ws += (size_t)BN * 2 * DH;  // ci, later c_in
  float* VU  = ws;  ws += (size_t)BN * DH;      // val_u
  float* VRC = ws;  ws += (size_t)BN * DH;      // val_r, later val_c
  float* UG  = ws;  ws += (size_t)BN * DH;      // u-gate
  float* RG  = T1;                              // reuse: T1 free after encoders

  const dim3 lb(32), lg(BN / 16);
  const dim3 gb(128), gg(NN / 64, NB);
  const dim3 eb(256), eg(BN * DH / 256);

  // Encoders
  lin_wmma_kernel<<<lg, lb, 0, stream>>>(x,  ei_w1, ei_b1, T1,   DIN, DH, ACT_RELU);
  lin_wmma_kernel<<<lg, lb, 0, stream>>>(T1, ei_w2, ei_b2, hinv, DH,  DH, ACT_NONE);
  lin_wmma_kernel<<<lg, lb, 0, stream>>>(x,  es_w1, es_b1, T1,   DIN, DH, ACT_RELU);
  lin_wmma_kernel<<<lg, lb, 0, stream>>>(T1, es_w2, es_b2, HS,   DH,  DH, ACT_NONE);
  // Environment classifier
  lin_wmma_kernel<<<lg, lb, 0, stream>>>(HS, env_w, env_b, envl, DH, NENV, ACT_NONE);
  // Causal graph factors
  lin_wmma_kernel<<<lg, lb, 0, stream>>>(hinv, g1_w, g1_b, Mm1, DH, DG, ACT_TANH);
  lin_wmma_kernel<<<lg, lb, 0, stream>>>(hinv, g2_w, g2_b, Mm2, DH, DG, ACT_TANH);
  // GRU input projections (ci = [h_inv, h_prev])
  concat_ci_kernel<<<eg, eb, 0, stream>>>(hinv, h_prev, CI);
  lin_wmma_kernel<<<lg, lb, 0, stream>>>(CI, wu_w, wu_b, VU,  2 * DH, DH, ACT_NONE);
  lin_wmma_kernel<<<lg, lb, 0, stream>>>(CI, wr_w, wr_b, VRC, 2 * DH, DH, ACT_NONE);
  // u, r gates with fused adj recomputation (async double-buffered LDS staging)
  gru_ur_kernel<<<gg, gb, 0, stream>>>(Mm1, Mm2, VU, VRC, UG, RG);
  // c_in = [h_inv, r*h_prev] (first half of CI already holds h_inv)
  cin_update_kernel<<<eg, eb, 0, stream>>>(RG, h_prev, CI);
  lin_wmma_kernel<<<lg, lb, 0, stream>>>(CI, wc_w, wc_b, VRC, 2 * DH, DH, ACT_NONE);
  // c + h_new with fused adj recomputation and GRU epilogue
  gru_c_kernel<<<gg, gb, 0, stream>>>(Mm1, Mm2, VRC, UG, h_prev, hnew);
  // prediction head
  lin_wmma_kernel<<<lg, lb, 0, stream>>>(hnew, pr_w, pr_b, pred, DH, DOUT, ACT_NONE);
}